// Network_68650757259518
// MI455X (gfx1250) — compile-verified
//
#include <hip/hip_runtime.h>

typedef __attribute__((ext_vector_type(2))) float v2f;
typedef __attribute__((ext_vector_type(8))) float v8f;

constexpr int BATCH = 8;
constexpr int NPTS  = 2048;
constexpr int BNP   = BATCH * NPTS;   // 16384 point columns
constexpr int KNN_K = 20;
constexpr int ANB   = 10;             // neighbors used by the bv/cos branch
constexpr int BNA   = BNP * ANB;      // 163840

__device__ inline float lrelu_f(float x) { return x > 0.f ? x : 0.2f * x; }

// ---------------------------------------------------------------------------
// WMMA helpers: fp32 16x16 tile via V_WMMA_F32_16X16X4_F32 (wave32).
// A: W[mb..mb+15][k], row-major, leading dim CI (compile-time).
// B: Xs[k][nb..nb+15], row-major, leading dim ldx (constant at call sites).
// CI templated so zero-pad guards fold away; A loaded as one b64.
// ---------------------------------------------------------------------------
template <int CI>
__device__ inline v8f wmma_accum(const float* __restrict__ W,
                                 const float* __restrict__ Xs, int ldx,
                                 int mb, int nb, int lane) {
  const int l = lane & 15;
  const int half = lane >> 4;
  const float* wrow = W + (size_t)(mb + l) * CI + 2 * half;
  const float* xcol = Xs + nb + l;
  constexpr int CI4 = (CI / 4) * 4;
  v8f acc = {};
#pragma unroll
  for (int kb = 0; kb < CI4; kb += 4) {
    const int k0 = kb + 2 * half;
    v2f a = *(const v2f*)(wrow + kb);          // W[m][k0], W[m][k0+1] (8B aligned: CI even)
    v2f bfr;
    bfr[0] = xcol[(size_t)k0 * ldx];
    bfr[1] = xcol[(size_t)(k0 + 1) * ldx];
    acc = __builtin_amdgcn_wmma_f32_16x16x4_f32(false, a, false, bfr,
                                                (short)0, acc, false, false);
  }
  if constexpr (CI % 4 == 2) {                 // ragged tail (CI = 98)
    v2f a = {};
    v2f bfr = {};
    if (half == 0) {                           // k = CI4, CI4+1 valid only in half 0
      a = *(const v2f*)(wrow + CI4);
      bfr[0] = xcol[(size_t)CI4 * ldx];
      bfr[1] = xcol[(size_t)(CI4 + 1) * ldx];
    }
    acc = __builtin_amdgcn_wmma_f32_16x16x4_f32(false, a, false, bfr,
                                                (short)0, acc, false, false);
  }
  return acc;
}

// Per-channel sum / sumsq of a 16x16 D tile. Lanes 0-15 & 16-31 hold the two
// 8-row halves (wave32 layout) -> xor-shuffles with mask<16 stay in-half.
__device__ inline void tile_stats(v8f acc, int mb, int lane, float* stats, int CO) {
  const int l = lane & 15;
  const int half = lane >> 4;
#pragma unroll
  for (int r = 0; r < 8; ++r) {
    float s = acc[r];
    float q = s * s;
#pragma unroll
    for (int off = 1; off < 16; off <<= 1) {
      s += __shfl_xor(s, off);
      q += __shfl_xor(q, off);
    }
    if (l == 0) {
      int m = mb + r + 8 * half;
      atomicAdd(&stats[m], s);
      atomicAdd(&stats[CO + m], q);
    }
  }
}

// Generic channel-major GEMM  Y[CO][NC] = W[CO][CI] * X[CI][NC].
// block = (32, CO/16), grid = NC/16. NC compile-time -> constant addressing.
template <int CO, int CI, int NC, bool STATS>
__global__ void gemm_ws(const float* __restrict__ W, const float* __restrict__ X,
                        float* __restrict__ Y, float* __restrict__ stats) {
  const int lane = threadIdx.x;
  const int mb = threadIdx.y * 16;
  const int nb = blockIdx.x * 16;
  v8f acc = wmma_accum<CI>(W, X, NC, mb, nb, lane);
  const int l = lane & 15;
  const int half = lane >> 4;
#pragma unroll
  for (int r = 0; r < 8; ++r)
    Y[(size_t)(mb + r + 8 * half) * NC + nb + l] = acc[r];
  if constexpr (STATS) tile_stats(acc, mb, lane, stats, CO);
}

// ---------------------------------------------------------------------------
// KNN: brute force 32-d, per-point sorted top-20 (ascending squared distance).
// ---------------------------------------------------------------------------
__global__ void knn_kernel(const float* __restrict__ co, int* __restrict__ idx20) {
  int p = blockIdx.x * blockDim.x + threadIdx.x;
  if (p >= BNP) return;
  int b = p >> 11, n = p & 2047;
  const float* base = co + (size_t)b * 32 * 2048;
  float self[32];
#pragma unroll
  for (int c = 0; c < 32; ++c) self[c] = base[(c << 11) + n];
  float d[KNN_K]; int di[KNN_K];
#pragma unroll
  for (int j = 0; j < KNN_K; ++j) { d[j] = 3.0e38f; di[j] = 0; }
  for (int m = 0; m < 2048; ++m) {
    __builtin_prefetch(base + m + 64, 0, 1);
    float dist = 0.f;
#pragma unroll
    for (int c = 0; c < 32; ++c) {
      float t = base[(c << 11) + m] - self[c];
      dist = fmaf(t, t, dist);
    }
    if (dist < d[KNN_K - 1]) {
      d[KNN_K - 1] = dist; di[KNN_K - 1] = m;
#pragma unroll
      for (int j = KNN_K - 1; j > 0; --j) {
        if (d[j] < d[j - 1]) {
          float td = d[j]; d[j] = d[j - 1]; d[j - 1] = td;
          int ti = di[j]; di[j] = di[j - 1]; di[j - 1] = ti;
        }
      }
    }
  }
#pragma unroll
  for (int j = 0; j < KNN_K; ++j) idx20[p * KNN_K + j] = di[j];
}

// (B,C,N) -> channel-major [C][BNP] with col = b*2048+n
__global__ void repack_kernel(const float* __restrict__ src, float* __restrict__ dst, int C) {
  int gid = blockIdx.x * 256 + threadIdx.x;
  if (gid >= C * BNP) return;
  int c = gid / BNP, col = gid % BNP;
  int b = col >> 11, n = col & 2047;
  dst[gid] = src[((size_t)(b * C + c) << 11) + n];
}

// diff[c][(p,j)] = co[idx]-co[self]; sv33 rows 0..31 = f_abs (gathered co)
__global__ void diff_kernel(const float* __restrict__ co, const int* __restrict__ idx20,
                            float* __restrict__ diff, float* __restrict__ sv33) {
  int pj = blockIdx.x * 256 + threadIdx.x;
  if (pj >= BNA) return;
  int p = pj / ANB, j = pj % ANB;
  int b = p >> 11, n = p & 2047;
  int id = idx20[p * KNN_K + 1 + j];  // drop self
  for (int c = 0; c < 32; ++c) {
    const float* row = co + ((size_t)(b * 32 + c) << 11);
    float fa = row[id];
    diff[(size_t)c * BNA + pj] = fa - row[n];
    sv33[(size_t)c * BNA + pj] = fa;
  }
}

// bn+lrelu on bv_pre, then mean over the 10 neighbors
__global__ void bv_mean_kernel(const float* __restrict__ bvpre, const float* __restrict__ sc,
                               const float* __restrict__ sh, float* __restrict__ bvm) {
  int gid = blockIdx.x * 256 + threadIdx.x;
  if (gid >= 32 * BNP) return;
  int c = gid / BNP, p = gid % BNP;
  float s = 0.f;
  for (int j = 0; j < ANB; ++j)
    s += lrelu_f(fmaf(sc[c], bvpre[(size_t)c * BNA + p * ANB + j], sh[c]));
  bvm[gid] = s * 0.1f;
}

__global__ void cos_kernel(const float* __restrict__ bvm, const float* __restrict__ diff,
                           float* __restrict__ cosb) {
  int pj = blockIdx.x * 256 + threadIdx.x;
  if (pj >= BNA) return;
  int p = pj / ANB;
  float inn = 0.f, d1 = 0.f, d2 = 0.f;
  for (int c = 0; c < 32; ++c) {
    float bv = bvm[(size_t)c * BNP + p];
    float df = diff[(size_t)c * BNA + pj];
    inn = fmaf(bv, df, inn);
    d1 = fmaf(bv, bv, d1);
    d2 = fmaf(df, df, d2);
  }
  cosb[pj] = inn * rsqrtf(d1 * d2 + 1e-10f);
}

// CO==1 conv (dot) + block-reduced stats
__global__ void dot_stats_kernel(const float* __restrict__ w, int CI,
                                 const float* __restrict__ X, float* __restrict__ Y,
                                 float* __restrict__ stats, int ncols) {
  int col = blockIdx.x * 256 + threadIdx.x;
  float s = 0.f;
  if (col < ncols)
    for (int c = 0; c < CI; ++c) s = fmaf(w[c], X[(size_t)c * ncols + col], s);
  if (col < ncols) Y[col] = s;
  __shared__ float r1[256], r2[256];
  r1[threadIdx.x] = (col < ncols) ? s : 0.f;
  r2[threadIdx.x] = (col < ncols) ? s * s : 0.f;
  __syncthreads();
  for (int o = 128; o > 0; o >>= 1) {
    if (threadIdx.x < o) { r1[threadIdx.x] += r1[threadIdx.x + o]; r2[threadIdx.x] += r2[threadIdx.x + o]; }
    __syncthreads();
  }
  if (threadIdx.x == 0) { atomicAdd(&stats[0], r1[0]); atomicAdd(&stats[1], r2[0]); }
}

__global__ void bn_finalize(const float* __restrict__ stats, const float* __restrict__ g,
                            const float* __restrict__ b, float* __restrict__ scale,
                            float* __restrict__ shift, int C, float inv_cnt) {
  int c = threadIdx.x;
  if (c >= C) return;
  float m = stats[c] * inv_cnt;
  float var = stats[C + c] * inv_cnt - m * m;
  float sc = g[c] * rsqrtf(var + 1e-5f);
  scale[c] = sc;
  shift[c] = b[c] - m * sc;
}

// ACT: 0 none, 1 relu, 2 lrelu; in-place on [C][ncols]
template <int ACT>
__global__ void bn_act_kernel(float* __restrict__ Y, const float* __restrict__ sc,
                              const float* __restrict__ sh, int ncols) {
  int gid = blockIdx.x * 256 + threadIdx.x;
  int c = gid / ncols;
  float v = fmaf(sc[c], Y[gid], sh[c]);
  if (ACT == 1) v = fmaxf(v, 0.f);
  if (ACT == 2) v = lrelu_f(v);
  Y[gid] = v;
}

// CO==1 bn+lrelu
__global__ void bn_act1_kernel(const float* __restrict__ Y, const float* __restrict__ sc,
                               const float* __restrict__ sh, float* __restrict__ out, int ncols) {
  int col = blockIdx.x * 256 + threadIdx.x;
  if (col >= ncols) return;
  out[col] = lrelu_f(fmaf(sc[0], Y[col], sh[0]));
}

// sv_sum: rows 64..97 of the f-stage input buffer
__global__ void svsum_kernel(const float* __restrict__ sv33, const float* __restrict__ t,
                             const float* __restrict__ cosb, float* __restrict__ fbuf) {
  int p = blockIdx.x * 256 + threadIdx.x;
  if (p >= BNP) return;
  for (int c = 0; c < 34; ++c) {
    float s = 0.f;
    for (int j = 0; j < ANB; ++j) {
      int pj = p * ANB + j;
      float sv = (c < 33) ? sv33[(size_t)c * BNA + pj] : t[pj];
      s = fmaf(cosb[pj], sv, s);
    }
    fbuf[(size_t)(64 + c) * BNP + p] = s;
  }
}

// np_max: rows 0..31 = max_k(pos), rows 32..95 = feature (max of constant)
__global__ void npmax_kernel(const float* __restrict__ co, const float* __restrict__ feature,
                             const int* __restrict__ idx20, float* __restrict__ npm) {
  int p = blockIdx.x * 256 + threadIdx.x;
  if (p >= BNP) return;
  int b = p >> 11, n = p & 2047;
  for (int c = 0; c < 32; ++c) {
    const float* row = co + ((size_t)(b * 32 + c) << 11);
    float selfv = row[n];
    float mx = -3.0e38f;
    for (int kk = 0; kk < KNN_K; ++kk)
      mx = fmaxf(mx, row[idx20[p * KNN_K + kk]] - selfv);
    npm[(size_t)c * BNP + p] = mx;
  }
  for (int c = 0; c < 64; ++c)
    npm[(size_t)(32 + c) * BNP + p] = feature[((size_t)(b * 64 + c) << 11) + n];
}

// Mqk = km_w^T * qm_w (collapses Qm/Km/Am/a1 sub-graph to rank-1)
__global__ void mqk_kernel(const float* __restrict__ km, const float* __restrict__ qm,
                           float* __restrict__ Mqk) {
  int gid = blockIdx.x * 256 + threadIdx.x;
  if (gid >= 64 * 64) return;
  int d = gid >> 6, e = gid & 63;
  float s = 0.f;
  for (int c = 0; c < 64; ++c) s = fmaf(km[c * 64 + d], qm[c * 64 + e], s);
  Mqk[gid] = s;
}

// ---------------------------------------------------------------------------
// Fused neighbor attention. One block (256 thr / 8 waves) per point; k padded
// 20->32 columns (pads are zeroed so BN stats stay exact).
// PHASE 0: p1 stats only.  PHASE 1: + a2 stats.  PHASE 2: full output.
// ---------------------------------------------------------------------------
template <int PHASE>
__launch_bounds__(256)
__global__ void attn_kernel(const float* __restrict__ co, const int* __restrict__ idx20,
                            const float* __restrict__ agg, const float* __restrict__ qs,
                            const float* __restrict__ Qb,
                            const float* __restrict__ p1w, const float* __restrict__ p2w,
                            const float* __restrict__ kvw, const float* __restrict__ a1w,
                            const float* __restrict__ a2w, const float* __restrict__ a3w,
                            const float* __restrict__ p1sc, const float* __restrict__ p1sh,
                            const float* __restrict__ a2sc, const float* __restrict__ a2sh,
                            float* __restrict__ p1stats, float* __restrict__ a2stats,
                            float* __restrict__ node1) {
  __shared__ float posL[32 * 32];
  __shared__ float nodeL[64 * 32];
  __shared__ float buf1[128 * 32];   // p1-relu -> a2-relu -> V
  __shared__ float buf2[128 * 32];   // pp
  __shared__ float bufA[128 * 32];   // K -> QK
  __shared__ float sL[32];
  __shared__ float Aw[32];
  __shared__ int nidx[KNN_K];

  const int p = blockIdx.x;
  const int b = p >> 11, n = p & 2047;
  const int tid = threadIdx.x;
  const int lane = tid & 31;
  const int wave = tid >> 5;
  const int l = lane & 15, half = lane >> 4;

  if (tid < KNN_K) nidx[tid] = idx20[p * KNN_K + tid];
  __syncthreads();

  for (int e = tid; e < 32 * 32; e += 256) {
    int c = e >> 5, kk = e & 31;
    float v = 0.f;
    if (kk < KNN_K) {
      const float* row = co + ((size_t)(b * 32 + c) << 11);
      v = row[nidx[kk]] - row[n];
    }
    posL[e] = v;
  }
  if constexpr (PHASE >= 1) {
    for (int e = tid; e < 64 * 32; e += 256) {
      int c = e >> 5, kk = e & 31;
      float v = 0.f;
      if (kk < KNN_K) v = agg[(size_t)c * BNP + (b << 11) + nidx[kk]];
      nodeL[e] = v;
    }
  }
  __syncthreads();

  // GEMM1: p1 (128x32) * pos
  for (int t = wave; t < 16; t += 8) {
    int mb = (t & 7) * 16, nb = (t >> 3) * 16;
    v8f acc = wmma_accum<32>(p1w, posL, 32, mb, nb, lane);
    if constexpr (PHASE == 0) {
      tile_stats(acc, mb, lane, p1stats, 128);
    } else {
#pragma unroll
      for (int r = 0; r < 8; ++r) {
        int m = mb + r + 8 * half;
        buf1[m * 32 + nb + l] = fmaxf(fmaf(p1sc[m], acc[r], p1sh[m]), 0.f);
      }
    }
  }
  if constexpr (PHASE == 0) return;
  __syncthreads();

  if (tid < 32) {  // s[k] = qs[:,p] . node[:,k]  (rank-1 "at" factor)
    float s = 0.f;
    if (tid < KNN_K)
      for (int c = 0; c < 64; ++c)
        s = fmaf(qs[(size_t)c * BNP + p], nodeL[c * 32 + tid], s);
    sL[tid] = s;
  }

  // GEMM2: pp = p2 (128x128) * relu(bn(p1.pos))
  for (int t = wave; t < 16; t += 8) {
    int mb = (t & 7) * 16, nb = (t >> 3) * 16;
    v8f acc = wmma_accum<128>(p2w, buf1, 32, mb, nb, lane);
#pragma unroll
    for (int r = 0; r < 8; ++r) buf2[(mb + r + 8 * half) * 32 + nb + l] = acc[r];
  }
  // GEMM3: K = kv[0:128] (128x64) * node
  for (int t = wave; t < 16; t += 8) {
    int mb = (t & 7) * 16, nb = (t >> 3) * 16;
    v8f acc = wmma_accum<64>(kvw, nodeL, 32, mb, nb, lane);
#pragma unroll
    for (int r = 0; r < 8; ++r) bufA[(mb + r + 8 * half) * 32 + nb + l] = acc[r];
  }
  __syncthreads();

  // QK = Q - K + pp + a1*s   (pad columns zeroed -> exact a2 stats)
  for (int e = tid; e < 128 * 32; e += 256) {
    int m = e >> 5, kk = e & 31;
    float v = 0.f;
    if (kk < KNN_K)
      v = Qb[(size_t)m * BNP + p] - bufA[e] + buf2[e] + a1w[m] * sL[kk];
    bufA[e] = v;
  }
  __syncthreads();

  // GEMM4: a2 (128x128) * QK
  for (int t = wave; t < 16; t += 8) {
    int mb = (t & 7) * 16, nb = (t >> 3) * 16;
    v8f acc = wmma_accum<128>(a2w, bufA, 32, mb, nb, lane);
    if constexpr (PHASE == 1) {
      tile_stats(acc, mb, lane, a2stats, 128);
    } else {
#pragma unroll
      for (int r = 0; r < 8; ++r) {
        int m = mb + r + 8 * half;
        buf1[m * 32 + nb + l] = fmaxf(fmaf(a2sc[m], acc[r], a2sh[m]), 0.f);
      }
    }
  }
  if constexpr (PHASE == 1) return;
  __syncthreads();

  // a3 logits + softmax over 20 real neighbors
  if (tid < 32) {
    float lg = 0.f;
    if (tid < KNN_K) {
      for (int m = 0; m < 128; ++m) lg = fmaf(a3w[m], buf1[m * 32 + tid], lg);
      lg *= 0.088388347648318447f;  // 1/sqrt(128)
    }
    Aw[tid] = lg;
  }
  __syncthreads();
  if (tid == 0) {
    float mx = -3.0e38f;
    for (int kk = 0; kk < KNN_K; ++kk) mx = fmaxf(mx, Aw[kk]);
    float ssum = 0.f;
    for (int kk = 0; kk < KNN_K; ++kk) { float e = __expf(Aw[kk] - mx); Aw[kk] = e; ssum += e; }
    float inv = 1.f / ssum;
    for (int kk = 0; kk < KNN_K; ++kk) Aw[kk] *= inv;
  }
  __syncthreads();

  // V = kv[128:256] * node  (reuse buf1)
  for (int t = wave; t < 16; t += 8) {
    int mb = (t & 7) * 16, nb = (t >> 3) * 16;
    v8f acc = wmma_accum<64>(kvw + 128 * 64, nodeL, 32, mb, nb, lane);
#pragma unroll
    for (int r = 0; r < 8; ++r) buf1[(mb + r + 8 * half) * 32 + nb + l] = acc[r];
  }
  __syncthreads();

  if (tid < 128) {
    float o = 0.f;
    for (int kk = 0; kk < KNN_K; ++kk)
      o = fmaf(Aw[kk], buf1[tid * 32 + kk] + buf2[tid * 32 + kk], o);
    node1[(size_t)tid * BNP + p] = o;
  }
}

// in-place A += B with per-channel stats ('h' stage); 256 | ncols so c uniform
__global__ void add_stats_kernel(float* __restrict__ A, const float* __restrict__ Bv,
                                 float* __restrict__ stats, int C, int ncols) {
  int gid = blockIdx.x * 256 + threadIdx.x;
  int c = gid / ncols;
  float v = A[gid] + Bv[gid];
  A[gid] = v;
  __shared__ float r1[256], r2[256];
  r1[threadIdx.x] = v;
  r2[threadIdx.x] = v * v;
  __syncthreads();
  for (int o = 128; o > 0; o >>= 1) {
    if (threadIdx.x < o) { r1[threadIdx.x] += r1[threadIdx.x + o]; r2[threadIdx.x] += r2[threadIdx.x + o]; }
    __syncthreads();
  }
  if (threadIdx.x == 0) { atomicAdd(&stats[c], r1[0]); atomicAdd(&stats[C + c], r2[0]); }
}

// bn + act + repack [C][BNP] -> (B,C,N) into d_out at `off`
template <int ACT>
__global__ void bn_act_out_kernel(const float* __restrict__ Y, const float* __restrict__ sc,
                                  const float* __restrict__ sh, float* __restrict__ out,
                                  int C, size_t off) {
  int gid = blockIdx.x * 256 + threadIdx.x;
  int c = gid / BNP, col = gid % BNP;
  int b = col >> 11, n = col & 2047;
  float v = fmaf(sc[c], Y[gid], sh[c]);
  if (ACT == 1) v = fmaxf(v, 0.f);
  if (ACT == 2) v = lrelu_f(v);
  out[off + ((size_t)(b * C + c) << 11) + n] = v;
}

// ---------------------------------------------------------------------------
extern "C" void kernel_launch(void* const* d_in, const int* in_sizes, int n_in,
                              void* d_out, int out_size, void* d_ws, size_t ws_size,
                              hipStream_t stream) {
  (void)in_sizes; (void)n_in; (void)out_size; (void)ws_size;
  const float* feature = (const float*)d_in[0];
  const float* co      = (const float*)d_in[1];
  const float* sk_w  = (const float*)d_in[4];
  const float* bv_w  = (const float*)d_in[5];
  const float* re_w  = (const float*)d_in[6];
  const float* se_w  = (const float*)d_in[7];
  const float* f_w   = (const float*)d_in[8];
  const float* qm_w  = (const float*)d_in[9];
  const float* km_w  = (const float*)d_in[10];
  const float* q_w   = (const float*)d_in[11];
  const float* kv_w  = (const float*)d_in[12];
  const float* a1_w  = (const float*)d_in[13];
  const float* a2_w  = (const float*)d_in[14];
  const float* a3_w  = (const float*)d_in[15];
  const float* p1_w  = (const float*)d_in[16];
  const float* p2_w  = (const float*)d_in[17];
  const float* conv_w = (const float*)d_in[18];
  const float* int_w  = (const float*)d_in[19];
  const float* sc_w   = (const float*)d_in[20];
  const float* sch_w  = (const float*)d_in[21];
  const float* co_w   = (const float*)d_in[22];
  // bn params in ref order: sk bv re se f a2 p1 conv int sc sch h co
  const float* g_[13]; const float* b_[13];
  for (int i = 0; i < 13; ++i) { g_[i] = (const float*)d_in[23 + 2 * i]; b_[i] = (const float*)d_in[24 + 2 * i]; }

  // ---- workspace carve (floats after idx ints) ----
  size_t o = 0;
  int* idx20 = (int*)d_ws;               o += (size_t)BNP * KNN_K;
  float* F = (float*)d_ws;
  float* stats  = F + o; o += 16 * 256;  // slot s -> stats + s*256 : [sum(C)|sumsq(C)]
  float* scaleA = F + o; o += 16 * 256;
  float* shiftA = F + o; o += 16 * 256;
  float* Mqk    = F + o; o += 64 * 64;
  float* diff   = F + o; o += (size_t)32 * BNA;
  float* sv33   = F + o; o += (size_t)33 * BNA;   // 0..31 f_abs, 32 diff_r
  float* bvpre  = F + o; o += (size_t)32 * BNA;
  float* bvm    = F + o; o += (size_t)32 * BNP;
  float* repre  = F + o; o += BNA;
  float* tpre   = F + o; o += BNA;
  float* cosb   = F + o; o += BNA;
  float* fbuf   = F + o; o += (size_t)98 * BNP;   // 0..63 feature, 64..97 sv_sum
  float* aggb   = F + o; o += (size_t)64 * BNP;
  float* npm    = F + o; o += (size_t)96 * BNP;
  float* cat    = F + o; o += (size_t)256 * BNP;  // 0..127 new_points, 128..255 conv-out
  float* qsb    = F + o; o += (size_t)64 * BNP;
  float* qbb    = F + o; o += (size_t)128 * BNP;
  float* n1pre  = F + o; o += (size_t)128 * BNP;
  float* selfb  = F + o; o += (size_t)128 * BNP;
  float* scbuf  = F + o; o += (size_t)128 * BNP;
  float* hbuf   = F + o; o += (size_t)128 * BNP;
  float* coT    = F + o; o += (size_t)32 * BNP;
  float* copre  = F + o; o += (size_t)64 * BNP;
  float* out = (float*)d_out;

  // stage slots: 0 bv,1 re,2 se,3 f,4 sk,5 p1,6 a2,7 conv,8 int,9 sc,10 sch,11 h,12 co
  auto ST = [&](int s) { return stats + s * 256; };
  auto SC = [&](int s) { return scaleA + s * 256; };
  auto SH = [&](int s) { return shiftA + s * 256; };
  hipMemsetAsync(stats, 0, 16 * 256 * sizeof(float), stream);

  // ---- KNN + repacks ----
  knn_kernel<<<BNP / 256, 256, 0, stream>>>(co, idx20);
  repack_kernel<<<(32 * BNP) / 256, 256, 0, stream>>>(co, coT, 32);
  repack_kernel<<<(64 * BNP) / 256, 256, 0, stream>>>(feature, fbuf, 64);
  diff_kernel<<<BNA / 256, 256, 0, stream>>>(co, idx20, diff, sv33);

  // ---- bv branch ----
  gemm_ws<32, 32, BNA, true><<<BNA / 16, dim3(32, 2), 0, stream>>>(bv_w, diff, bvpre, ST(0));
  bn_finalize<<<1, 32, 0, stream>>>(ST(0), g_[1], b_[1], SC(0), SH(0), 32, 1.f / BNA);
  bv_mean_kernel<<<(32 * BNP) / 256, 256, 0, stream>>>(bvpre, SC(0), SH(0), bvm);
  cos_kernel<<<BNA / 256, 256, 0, stream>>>(bvm, diff, cosb);

  // ---- re / se (CO==1) ----
  dot_stats_kernel<<<BNA / 256, 256, 0, stream>>>(re_w, 32, diff, repre, ST(1), BNA);
  bn_finalize<<<1, 1, 0, stream>>>(ST(1), g_[2], b_[2], SC(1), SH(1), 1, 1.f / BNA);
  bn_act1_kernel<<<BNA / 256, 256, 0, stream>>>(repre, SC(1), SH(1), sv33 + (size_t)32 * BNA, BNA);
  dot_stats_kernel<<<BNA / 256, 256, 0, stream>>>(se_w, 33, sv33, tpre, ST(2), BNA);
  bn_finalize<<<1, 1, 0, stream>>>(ST(2), g_[3], b_[3], SC(2), SH(2), 1, 1.f / BNA);
  bn_act1_kernel<<<BNA / 256, 256, 0, stream>>>(tpre, SC(2), SH(2), tpre, BNA);
  svsum_kernel<<<BNP / 256, 256, 0, stream>>>(sv33, tpre, cosb, fbuf);

  // ---- f stage -> agg_f ----
  gemm_ws<64, 98, BNP, true><<<BNP / 16, dim3(32, 4), 0, stream>>>(f_w, fbuf, aggb, ST(3));
  bn_finalize<<<1, 64, 0, stream>>>(ST(3), g_[4], b_[4], SC(3), SH(3), 64, 1.f / BNP);
  bn_act_kernel<1><<<(64 * BNP) / 256, 256, 0, stream>>>(aggb, SC(3), SH(3), BNP);

  // ---- skeleton branch -> new_points (cat rows 0..127) ----
  npmax_kernel<<<BNP / 256, 256, 0, stream>>>(co, feature, idx20, npm);
  gemm_ws<128, 96, BNP, true><<<BNP / 16, dim3(32, 8), 0, stream>>>(sk_w, npm, cat, ST(4));
  bn_finalize<<<1, 128, 0, stream>>>(ST(4), g_[0], b_[0], SC(4), SH(4), 128, 1.f / BNP);
  bn_act_kernel<1><<<(128 * BNP) / 256, 256, 0, stream>>>(cat, SC(4), SH(4), BNP);

  // ---- attention precomputes ----
  mqk_kernel<<<16, 256, 0, stream>>>(km_w, qm_w, Mqk);
  gemm_ws<64, 64, BNP, false><<<BNP / 16, dim3(32, 4), 0, stream>>>(Mqk, aggb, qsb, nullptr);
  gemm_ws<128, 64, BNP, false><<<BNP / 16, dim3(32, 8), 0, stream>>>(q_w, aggb, qbb, nullptr);

  // ---- fused attention: stats pass p1, stats pass a2, then full ----
  attn_kernel<0><<<BNP, 256, 0, stream>>>(co, idx20, aggb, qsb, qbb, p1_w, p2_w, kv_w,
      a1_w, a2_w, a3_w, SC(5), SH(5), SC(6), SH(6), ST(5), ST(6), n1pre);
  bn_finalize<<<1, 128, 0, stream>>>(ST(5), g_[6], b_[6], SC(5), SH(5), 128, 1.f / (BNP * KNN_K));
  attn_kernel<1><<<BNP, 256, 0, stream>>>(co, idx20, aggb, qsb, qbb, p1_w, p2_w, kv_w,
      a1_w, a2_w, a3_w, SC(5), SH(5), SC(6), SH(6), ST(5), ST(6), n1pre);
  bn_finalize<<<1, 128, 0, stream>>>(ST(6), g_[5], b_[5], SC(6), SH(6), 128, 1.f / (BNP * KNN_K));
  attn_kernel<2><<<BNP, 256, 0, stream>>>(co, idx20, aggb, qsb, qbb, p1_w, p2_w, kv_w,
      a1_w, a2_w, a3_w, SC(5), SH(5), SC(6), SH(6), ST(5), ST(6), n1pre);

  // ---- conv stage -> cat rows 128..255 ----
  gemm_ws<128, 128, BNP, true><<<BNP / 16, dim3(32, 8), 0, stream>>>(conv_w, n1pre, cat + (size_t)128 * BNP, ST(7));
  bn_finalize<<<1, 128, 0, stream>>>(ST(7), g_[7], b_[7], SC(7), SH(7), 128, 1.f / BNP);
  bn_act_kernel<2><<<(128 * BNP) / 256, 256, 0, stream>>>(cat + (size_t)128 * BNP, SC(7), SH(7), BNP);

  // ---- int / sc / sch / h ----
  gemm_ws<128, 256, BNP, true><<<BNP / 16, dim3(32, 8), 0, stream>>>(int_w, cat, selfb, ST(8));
  bn_finalize<<<1, 128, 0, stream>>>(ST(8), g_[8], b_[8], SC(8), SH(8), 128, 1.f / BNP);
  bn_act_kernel<2><<<(128 * BNP) / 256, 256, 0, stream>>>(selfb, SC(8), SH(8), BNP);

  gemm_ws<128, 256, BNP, true><<<BNP / 16, dim3(32, 8), 0, stream>>>(sc_w, cat, scbuf, ST(9));
  bn_finalize<<<1, 128, 0, stream>>>(ST(9), g_[9], b_[9], SC(9), SH(9), 128, 1.f / BNP);
  bn_act_kernel<0><<<(128 * BNP) / 256, 256, 0, stream>>>(scbuf, SC(9), SH(9), BNP);

  gemm_ws<128, 128, BNP, true><<<BNP / 16, dim3(32, 8), 0, stream>>>(sch_w, selfb, hbuf, ST(10));
  bn_finalize<<<1, 128, 0, stream>>>(ST(10), g_[10], b_[10], SC(10), SH(10), 128, 1.f / BNP);
  bn_act_kernel<0><<<(128 * BNP) / 256, 256, 0, stream>>>(hbuf, SC(10), SH(10), BNP);

  add_stats_kernel<<<(128 * BNP) / 256, 256, 0, stream>>>(scbuf, hbuf, ST(11), 128, BNP);
  bn_finalize<<<1, 128, 0, stream>>>(ST(11), g_[11], b_[11], SC(11), SH(11), 128, 1.f / BNP);
  bn_act_out_kernel<2><<<(128 * BNP) / 256, 256, 0, stream>>>(scbuf, SC(11), SH(11), out, 128, 0);

  // ---- co_out ----
  gemm_ws<64, 32, BNP, true><<<BNP / 16, dim3(32, 4), 0, stream>>>(co_w, coT, copre, ST(12));
  bn_finalize<<<1, 64, 0, stream>>>(ST(12), g_[12], b_[12], SC(12), SH(12), 64, 1.f / BNP);
  bn_act_out_kernel<1><<<(64 * BNP) / 256, 256, 0, stream>>>(copre, SC(12), SH(12), out, 64, (size_t)128 * BNP);
}